// ProbHypernet_27504970564166
// MI455X (gfx1250) — compile-verified
//
#include <hip/hip_runtime.h>
#include <cstdint>
#include <cstddef>

#define MB   128
#define IND  1024
#define OUTD 1024
#define HD   64
#define H1   65
#define RD   1025

typedef __attribute__((ext_vector_type(2))) float v2f;
typedef __attribute__((ext_vector_type(4))) float v4f;
typedef __attribute__((ext_vector_type(8))) float v8f;

// ---------------- Kernel 1: h = tanh(vmg_linear(x)) -> ha (128 x 65, bias col) ----
__global__ __launch_bounds__(64) void k_hidden(
    const float* __restrict__ x, const float* __restrict__ mu_xh,
    const float* __restrict__ lvr_xh, const float* __restrict__ lvc_xh,
    const float* __restrict__ E_xh, float* __restrict__ ha)
{
    __shared__ float s_xa[RD];   // xa[m, r]
    __shared__ float s_sr[RD];   // xa[m, r] * exp(0.5*lvr[r])
    const int m = blockIdx.x;
    const int t = threadIdx.x;
    for (int r = t; r < RD; r += 64) {
        float xv = (r < IND) ? x[(size_t)m * IND + r] : 1.0f;
        s_xa[r] = xv;
        s_sr[r] = xv * __expf(0.5f * lvr_xh[r]);
    }
    __syncthreads();

    const int o = t;  // 64 threads == HD outputs
    float accMu = 0.0f, accE = 0.0f;
    const float* Ep = E_xh + (size_t)m * RD * HD + o;
    for (int r = 0; r < RD; ++r) {
        float e = __builtin_nontemporal_load(Ep + (size_t)r * HD);
        accMu = fmaf(s_xa[r], mu_xh[r * HD + o], accMu);
        accE  = fmaf(s_sr[r], e, accE);
    }
    float sc = __expf(0.5f * lvc_xh[o]);
    ha[m * H1 + o] = tanhf(accMu + sc * accE);
    if (t == 0) ha[m * H1 + HD] = 1.0f;  // bias column
}

// ---------------- Kernel 2: second-stage vmg_linear with fused output transform ---
// mode 0: out = xa * val            (A = xa * mu_scaling,  J = RD)
// mode 1: out = xa * exp(0.5*val)   (B = xa * exp(.5 lvr), J = RD)
// mode 2: out = exp(0.5*val)        (SC = exp(.5 lvc),     J = OUTD)
__global__ __launch_bounds__(256) void k_stage2(
    const float* __restrict__ ha, const float* __restrict__ mu,
    const float* __restrict__ lvr, const float* __restrict__ lvc,
    const float* __restrict__ E, const float* __restrict__ x,
    float* __restrict__ outb, int J, int mode)
{
    __shared__ float s_ha[H1];
    __shared__ float s_hb[H1];
    const int m = blockIdx.y;
    const int t = threadIdx.x;
    if (t < H1) {
        float hv = ha[m * H1 + t];
        s_ha[t] = hv;
        s_hb[t] = hv * __expf(0.5f * lvr[t]);
    }
    __syncthreads();

    const int j = blockIdx.x * 256 + t;
    if (j >= J) return;

    float accMu = 0.0f, accE = 0.0f;
    const float* Ep = E + (size_t)m * H1 * J + j;
    #pragma unroll 5
    for (int k = 0; k < H1; ++k) {
        accMu = fmaf(s_ha[k], mu[(size_t)k * J + j], accMu);
        accE  = fmaf(s_hb[k], __builtin_nontemporal_load(Ep + (size_t)k * J), accE);
    }
    float val = accMu + __expf(0.5f * lvc[j]) * accE;
    if (mode == 0) {
        float xa = (j < IND) ? x[(size_t)m * IND + j] : 1.0f;
        outb[(size_t)m * J + j] = xa * val;
    } else if (mode == 1) {
        float xa = (j < IND) ? x[(size_t)m * IND + j] : 1.0f;
        outb[(size_t)m * J + j] = xa * __expf(0.5f * val);
    } else {
        outb[(size_t)m * J + j] = __expf(0.5f * val);
    }
}

// ---------------- Kernel 3: T1 = A(128x1025) @ M(1025x1024) via WMMA f32 ----------
// One 16x16 tile per wave; K loop over 1024 in steps of 4, scalar tail k=1024.
__global__ __launch_bounds__(128) void k_gemm_t1(
    const float* __restrict__ A, const float* __restrict__ Mx,
    float* __restrict__ T1)
{
    const int tid  = threadIdx.x;
    const int wave = tid >> 5;
    const int lane = tid & 31;
    const int ln   = lane & 15;
    const int half = lane >> 4;
    const int rowBase = blockIdx.y * 16;
    const int colBase = blockIdx.x * 64 + wave * 16;

    v8f acc = {0.f, 0.f, 0.f, 0.f, 0.f, 0.f, 0.f, 0.f};
    const float* Arow = A + (size_t)(rowBase + ln) * RD;

    for (int kb = 0; kb < IND; kb += 4) {
        const int k0 = kb + 2 * half;
        v2f a, b;
        // A 16x4 layout: lane L, vgpr v -> A[L%16][2*(L/16)+v]
        a.x = Arow[k0];
        a.y = Arow[k0 + 1];
        // B 4x16 layout: lane L, vgpr v -> B[2*(L/16)+v][L%16]
        b.x = Mx[(size_t)k0 * OUTD + colBase + ln];
        b.y = Mx[(size_t)(k0 + 1) * OUTD + colBase + ln];
        acc = __builtin_amdgcn_wmma_f32_16x16x4_f32(
            false, a, false, b, (short)0, acc, false, false);
    }

    // K tail (k = 1024 == IND) + store. D layout: lane L, vgpr v -> D[v+8*(L/16)][L%16]
    const int col = colBase + ln;
    const float mtail = Mx[(size_t)IND * OUTD + col];
    #pragma unroll
    for (int v = 0; v < 8; ++v) {
        const int row = rowBase + v + 8 * half;
        float d = acc[v] + A[(size_t)row * RD + IND] * mtail;
        T1[(size_t)row * OUTD + col] = d;
    }
}

// ---------------- Kernel 4: out = T1 + SC * (B-row . E_w)  (streams 537 MB) -------
__global__ __launch_bounds__(64) void k_final(
    const float* __restrict__ Bmr, const float* __restrict__ SC,
    const float* __restrict__ T1, const float* __restrict__ E_w,
    float* __restrict__ out)
{
    __shared__ float s_b[RD];
    const int m = blockIdx.y;
    const int t = threadIdx.x;
    for (int r = t; r < RD; r += 64) s_b[r] = Bmr[(size_t)m * RD + r];
    __syncthreads();

    const int obase = blockIdx.x * 256 + t * 4;
    const float* Ep = E_w + (size_t)m * RD * OUTD + obase;
    v4f acc = {0.f, 0.f, 0.f, 0.f};
    #pragma unroll 4
    for (int r = 0; r < RD; ++r) {
        v4f e = __builtin_nontemporal_load((const v4f*)(Ep + (size_t)r * OUTD));
        acc += s_b[r] * e;  // 4x FMA on a b128 NT stream
    }
    const size_t oi = (size_t)m * OUTD + obase;
    v4f t1 = *(const v4f*)(T1 + oi);
    v4f sc = *(const v4f*)(SC + oi);
    v4f res = t1 + sc * acc;
    *(v4f*)(out + oi) = res;
}

// ---------------- Kernel 5: KL scalar ---------------------------------------------
__device__ float blockReduce256(float v, float* s) {
    const int t = threadIdx.x;
    s[t] = v; __syncthreads();
    for (int off = 128; off > 0; off >>= 1) {
        if (t < off) s[t] += s[t + off];
        __syncthreads();
    }
    float r = s[0];
    __syncthreads();
    return r;
}

__global__ __launch_bounds__(256) void k_kl(
    const float* __restrict__ mu1, const float* __restrict__ lvr1, const float* __restrict__ lvc1,
    const float* __restrict__ mu2, const float* __restrict__ lvr2, const float* __restrict__ lvc2,
    const float* __restrict__ mu3, const float* __restrict__ lvr3, const float* __restrict__ lvc3,
    const float* __restrict__ mu4, const float* __restrict__ lvr4, const float* __restrict__ lvc4,
    float* __restrict__ out_kl)
{
    __shared__ float s[256];
    const int t = threadIdx.x;
    const float* mus[4]  = {mu1, mu2, mu3, mu4};
    const float* lvrs[4] = {lvr1, lvr2, lvr3, lvr4};
    const float* lvcs[4] = {lvc1, lvc2, lvc3, lvc4};
    const int rs[4] = {RD, H1, H1, H1};
    const int cs[4] = {HD, RD, RD, OUTD};
    float total = 0.0f;
    for (int L = 0; L < 4; ++L) {
        const int r = rs[L], c = cs[L];
        float pvr = 0.f, plr = 0.f, pvc = 0.f, plc = 0.f, pm2 = 0.f;
        for (int i = t; i < r; i += 256) { float lv = lvrs[L][i]; pvr += __expf(lv); plr += lv; }
        for (int i = t; i < c; i += 256) { float lv = lvcs[L][i]; pvc += __expf(lv); plc += lv; }
        const int n = r * c;
        for (int i = t; i < n; i += 256) { float mv = mus[L][i]; pm2 = fmaf(mv, mv, pm2); }
        float Svr = blockReduce256(pvr, s);
        float Slr = blockReduce256(plr, s);
        float Svc = blockReduce256(pvc, s);
        float Slc = blockReduce256(plc, s);
        float Sm2 = blockReduce256(pm2, s);
        total += 0.5f * (Svr * Svc + Sm2 - (float)r * (float)c
                         - (float)c * Slr - (float)r * Slc);
    }
    if (t == 0) *out_kl = total;
}

// ---------------- Launch ----------------------------------------------------------
extern "C" void kernel_launch(void* const* d_in, const int* in_sizes, int n_in,
                              void* d_out, int out_size, void* d_ws, size_t ws_size,
                              hipStream_t stream) {
    const float* x       = (const float*)d_in[0];
    const float* Mx      = (const float*)d_in[1];
    const float* mu_xh   = (const float*)d_in[2];
    const float* lvr_xh  = (const float*)d_in[3];
    const float* lvc_xh  = (const float*)d_in[4];
    const float* mu_hmu  = (const float*)d_in[5];
    const float* lvr_hmu = (const float*)d_in[6];
    const float* lvc_hmu = (const float*)d_in[7];
    const float* mu_hli  = (const float*)d_in[8];
    const float* lvr_hli = (const float*)d_in[9];
    const float* lvc_hli = (const float*)d_in[10];
    const float* mu_hlo  = (const float*)d_in[11];
    const float* lvr_hlo = (const float*)d_in[12];
    const float* lvc_hlo = (const float*)d_in[13];
    const float* E_xh    = (const float*)d_in[14];
    const float* E_hmu   = (const float*)d_in[15];
    const float* E_hli   = (const float*)d_in[16];
    const float* E_hlo   = (const float*)d_in[17];
    const float* E_w     = (const float*)d_in[18];

    float* ws  = (float*)d_ws;
    float* ha  = ws;                 // 128*65
    float* Amr = ha  + MB * H1;      // 128*1025
    float* Bmr = Amr + MB * RD;      // 128*1025
    float* SC  = Bmr + MB * RD;      // 128*1024
    float* T1  = SC  + MB * OUTD;    // 128*1024   (~2.1 MB total)
    float* out = (float*)d_out;      // 128*1024 + 1

    k_hidden<<<MB, 64, 0, stream>>>(x, mu_xh, lvr_xh, lvc_xh, E_xh, ha);

    dim3 g2r((RD + 255) / 256, MB);
    dim3 g2c((OUTD + 255) / 256, MB);
    k_stage2<<<g2r, 256, 0, stream>>>(ha, mu_hmu, lvr_hmu, lvc_hmu, E_hmu, x, Amr, RD, 0);
    k_stage2<<<g2r, 256, 0, stream>>>(ha, mu_hli, lvr_hli, lvc_hli, E_hli, x, Bmr, RD, 1);
    k_stage2<<<g2c, 256, 0, stream>>>(ha, mu_hlo, lvr_hlo, lvc_hlo, E_hlo, x, SC, OUTD, 2);

    k_gemm_t1<<<dim3(OUTD / 64, MB / 16), 128, 0, stream>>>(Amr, Mx, T1);

    k_final<<<dim3(OUTD / 256, MB), 64, 0, stream>>>(Bmr, SC, T1, E_w, out);

    k_kl<<<1, 256, 0, stream>>>(mu_xh, lvr_xh, lvc_xh, mu_hmu, lvr_hmu, lvc_hmu,
                                mu_hli, lvr_hli, lvc_hli, mu_hlo, lvr_hlo, lvc_hlo,
                                out + (size_t)MB * OUTD);
}